// EIGM_41197326303431
// MI455X (gfx1250) — compile-verified
//
#include <hip/hip_runtime.h>
#include <math.h>

#define Hdim 2048
#define Wdim 2048
#define HWSZ (Hdim * Wdim)
#define NSEG 10
#define NB 1024

// ws word layout
#define W_MIN 0
#define W_MAX 1
#define W_SEGSUM 2                    // 16x16 f32 (row M*16+N); col3 = counts
#define W_HCNT (W_SEGSUM + 256)       // NSEG*NB u32
#define W_HSUM (W_HCNT + NSEG * NB)   // NSEG*NB f32
#define W_DSEG (W_HSUM + NSEG * NB)   // NSEG*3 f32
#define W_PAR (W_DSEG + NSEG * 3)     // dmin, dmax, segscale
#define W_TOTAL (W_PAR + 3)

typedef __attribute__((ext_vector_type(2))) float v2f;
typedef __attribute__((ext_vector_type(8))) float v8f;

static __device__ __forceinline__ void wait_async0() {
#if __has_builtin(__builtin_amdgcn_s_wait_asynccnt)
    __builtin_amdgcn_s_wait_asynccnt(0);
#else
    asm volatile("s_wait_asynccnt 0" ::: "memory");
#endif
}

__global__ __launch_bounds__(256) void eigm_init(unsigned int* ws) {
    int i = blockIdx.x * 256 + threadIdx.x;
    if (i < W_TOTAL) ws[i] = 0u;
    if (i == 0) ws[W_MIN] = 0x7F800000u;  // +inf bits
}

__global__ __launch_bounds__(256) void eigm_minmax(const float* __restrict__ depth,
                                                   unsigned int* ws) {
    __shared__ float smin[256];
    __shared__ float smax[256];
    int tid = threadIdx.x;
    float mn = INFINITY, mx = 0.0f;
    for (int i = blockIdx.x * 256 + tid; i < HWSZ; i += gridDim.x * 256) {
        float d = depth[i];
        mn = fminf(mn, d);
        mx = fmaxf(mx, d);
    }
    smin[tid] = mn; smax[tid] = mx;
    __syncthreads();
    for (int o = 128; o > 0; o >>= 1) {
        if (tid < o) {
            smin[tid] = fminf(smin[tid], smin[tid + o]);
            smax[tid] = fmaxf(smax[tid], smax[tid + o]);
        }
        __syncthreads();
    }
    if (tid == 0) {
        atomicMin(&ws[W_MIN], __float_as_uint(smin[0]));  // depth > 0: uint order == float order
        atomicMax(&ws[W_MAX], __float_as_uint(smax[0]));
    }
}

// Segment stats via V_WMMA_F32_16X16X4_F32:
//   A(16x4) = onehot(seg(pixel_k) == m), B(4x16) = [r,g,b,1,0...] per pixel
//   C[m][c] += per-segment channel sums; C[m][3] = segment counts (free).
// Wave stages its 32 pixels as {r,g,b,1,segf} (stride 5 dwords) in LDS; each
// lane then reads only the 4 dwords it needs per WMMA step (ds_load_2addr
// friendly) instead of 8 ds_bpermute broadcasts.
__global__ __launch_bounds__(256) void eigm_accum(const float* __restrict__ img,
                                                  const float* __restrict__ depth,
                                                  unsigned int* wsU) {
    __shared__ unsigned int lCnt[NSEG * NB];
    __shared__ float lSum[NSEG * NB];
    __shared__ float stage[8 * 32 * 5];
    float* wsF = (float*)wsU;

    int tid = threadIdx.x;
    for (int i = tid; i < NSEG * NB; i += 256) { lCnt[i] = 0u; lSum[i] = 0.0f; }
    __syncthreads();

    float dmin = wsF[W_MIN];
    float dmax = wsF[W_MAX];
    float diff = dmax - dmin;
    float scale = (diff > 0.0f) ? ((float)NSEG / diff) : 0.0f;

    const float* __restrict__ imgR = img;
    const float* __restrict__ imgG = img + HWSZ;
    const float* __restrict__ imgB = img + 2 * HWSZ;

    int lane = tid & 31;
    int wid = tid >> 5;
    int gw = blockIdx.x * 8 + wid;
    int totW = gridDim.x * 8;
    int m = lane & 15;
    int kk = (lane < 16) ? 0 : 2;
    float mf = (float)m;
    int nsel = (m < 3) ? m : 3;
    bool useb = (m < 4);

    float* st = &stage[(wid * 32 + lane) * 5];
    const float* wb = &stage[wid * 32 * 5];

    v8f acc = {0.f, 0.f, 0.f, 0.f, 0.f, 0.f, 0.f, 0.f};

    for (int base = gw * 32; base < HWSZ; base += totW * 32) {
        int pix = base + lane;
        float d = depth[pix];
        float r = imgR[pix];
        float g = imgG[pix];
        float bl = imgB[pix];
        int sg;
        if (d < dmax) {
            int t = (int)((d - dmin) * scale);
            sg = (t > NSEG - 1) ? (NSEG - 1) : t;
        } else {
            sg = -1;  // exactly-max pixel belongs to no bin (d < bins[-1] fails)
        }
        float mnrgb = fminf(r, fminf(g, bl));
        if (sg >= 0) {
            int bin = (int)(mnrgb * (float)NB);
            if (bin > NB - 1) bin = NB - 1;
            if (bin < 0) bin = 0;
            atomicAdd(&lCnt[sg * NB + bin], 1u);
            atomicAdd(&lSum[sg * NB + bin], mnrgb);
        }
        st[0] = r; st[1] = g; st[2] = bl; st[3] = 1.0f; st[4] = (float)sg;
        __builtin_amdgcn_wave_barrier();  // within-wave LDS is in-order; fence the compiler
#pragma unroll
        for (int j = 0; j < 8; ++j) {
            const float* pp = wb + (4 * j + kk) * 5;
            float v0 = pp[nsel];
            float v1 = pp[5 + nsel];
            float s0 = pp[4];
            float s1 = pp[9];
            v2f va, vbv;
            va.x = (s0 == mf) ? 1.0f : 0.0f;
            va.y = (s1 == mf) ? 1.0f : 0.0f;
            vbv.x = useb ? v0 : 0.0f;
            vbv.y = useb ? v1 : 0.0f;
            acc = __builtin_amdgcn_wmma_f32_16x16x4_f32(false, va, false, vbv,
                                                        (short)0, acc, false, false);
        }
        __builtin_amdgcn_wave_barrier();  // don't let next iter's stores slide above reads
    }

    // C layout: VGPR j -> M = j + (lane<16?0:8), N = lane&15
#pragma unroll
    for (int j = 0; j < 8; ++j) {
        int M = j + ((lane < 16) ? 0 : 8);
        atomicAdd(&wsF[W_SEGSUM + M * 16 + m], acc[j]);
    }

    __syncthreads();
    for (int i = tid; i < NSEG * NB; i += 256) {
        if (lCnt[i]) {
            atomicAdd(&wsU[W_HCNT + i], lCnt[i]);
            atomicAdd(&wsF[W_HSUM + i], lSum[i]);
        }
    }
}

__global__ void eigm_stats(unsigned int* wsU) {
    float* wsF = (float*)wsU;
    int s = threadIdx.x;
    float dmin = wsF[W_MIN];
    float dmax = wsF[W_MAX];
    if (s == 0) {
        wsF[W_PAR + 0] = dmin;
        wsF[W_PAR + 1] = dmax;
        float diff = dmax - dmin;
        wsF[W_PAR + 2] = (diff > 0.0f) ? ((float)NSEG / diff) : 0.0f;
    }
    if (s < NSEG) {
        float nf = wsF[W_SEGSUM + s * 16 + 3];
        long long n = (long long)(nf + 0.5f);
        long long k = n / 100;  // BOTTOM_PCT = 1
        float Bc = 0.0f;
        if (k > 0) {
            long long cum = 0;
            float sumB = 0.0f;
            for (int bin = 0; bin < NB; ++bin) {
                unsigned int c = wsU[W_HCNT + s * NB + bin];
                float ss = wsF[W_HSUM + s * NB + bin];
                if (cum + (long long)c >= k) {
                    float need = (float)(k - cum);
                    float avg = (c > 0) ? (ss / (float)c) : 0.0f;
                    sumB += need * avg;
                    break;
                }
                cum += c;
                sumB += ss;
            }
            Bc = sumB / (float)k;
        }
        float invn = (nf > 0.0f) ? (1.0f / nf) : 0.0f;
        wsF[W_DSEG + s * 3 + 0] = wsF[W_SEGSUM + s * 16 + 0] * invn - Bc;
        wsF[W_DSEG + s * 3 + 1] = wsF[W_SEGSUM + s * 16 + 1] * invn - Bc;
        wsF[W_DSEG + s * 3 + 2] = wsF[W_SEGSUM + s * 16 + 2] * invn - Bc;
    }
}

// Fused: D scatter + 3x3 depth-validity average + d(x) scaling. J = (D + a') * d
// since F_SCALE*P_MIX = F_SCALE*(1-P_MIX) = 1.
// Halo depth tile is brought in with CDNA5 async-to-LDS loads (ASYNCcnt path);
// OOB cells use clamped addresses, then are rewritten to +inf after the wait.
__global__ __launch_bounds__(256) void eigm_final(const float* __restrict__ img,
                                                  const float* __restrict__ depth,
                                                  const float* __restrict__ mu0,
                                                  const float* __restrict__ mu1,
                                                  const float* __restrict__ mu2,
                                                  const float* __restrict__ wsF,
                                                  float* __restrict__ out) {
    __shared__ float sd[10 * 34];
    __shared__ int ssg[10 * 34];
    __shared__ float sD[NSEG * 3];

    int tx = threadIdx.x, ty = threadIdx.y;
    int tid = ty * 32 + tx;
    float dmin = wsF[W_PAR + 0];
    float dmax = wsF[W_PAR + 1];
    float scale = wsF[W_PAR + 2];

    // Phase 1: async halo fetch (coordinates clamped into the image)
    for (int i = tid; i < 34 * 10; i += 256) {
        int lx = i % 34, ly = i / 34;
        int x = blockIdx.x * 32 + lx - 1;
        int y = blockIdx.y * 8 + ly - 1;
        int xc = (x < 0) ? 0 : ((x > Wdim - 1) ? (Wdim - 1) : x);
        int yc = (y < 0) ? 0 : ((y > Hdim - 1) ? (Hdim - 1) : y);
        unsigned goff = (unsigned)((yc * Wdim + xc) * 4);
        unsigned ldsa = (unsigned)(size_t)&sd[i];
        asm volatile("global_load_async_to_lds_b32 %0, %1, %2"
                     :
                     : "v"(ldsa), "v"(goff), "s"(depth)
                     : "memory");
    }
    wait_async0();
    __syncthreads();

    // Phase 2: fix OOB cells to +inf, derive segment ids
    for (int i = tid; i < 34 * 10; i += 256) {
        int lx = i % 34, ly = i / 34;
        int x = blockIdx.x * 32 + lx - 1;
        int y = blockIdx.y * 8 + ly - 1;
        float d = sd[i];
        bool in = (x >= 0) && (x < Wdim) && (y >= 0) && (y < Hdim);
        if (!in) { d = INFINITY; sd[i] = d; }
        int sg = -1;
        if (d < dmax) {
            int t = (int)((d - dmin) * scale);
            sg = (t > NSEG - 1) ? (NSEG - 1) : t;
        }
        ssg[i] = sg;
    }
    if (tid < NSEG * 3) sD[tid] = wsF[W_DSEG + tid];
    __syncthreads();

    int gx = blockIdx.x * 32 + tx;
    int gy = blockIdx.y * 8 + ty;
    int idx = gy * Wdim + gx;

    float dc = sd[(ty + 1) * 34 + tx + 1];
    int sc = ssg[(ty + 1) * 34 + tx + 1];
    float n0 = 0.f, n1 = 0.f, n2 = 0.f, cnt = 0.f;
#pragma unroll
    for (int dy = 0; dy < 3; ++dy) {
#pragma unroll
        for (int dx = 0; dx < 3; ++dx) {
            float nd = sd[(ty + dy) * 34 + tx + dx];
            if (fabsf(nd - dc) < 1.0f) {
                cnt += 1.0f;
                int ns = ssg[(ty + dy) * 34 + tx + dx];
                if (ns >= 0) {
                    n0 += sD[ns * 3 + 0];
                    n1 += sD[ns * 3 + 1];
                    n2 += sD[ns * 3 + 2];
                }
            }
        }
    }
    float r = img[idx], g = img[HWSZ + idx], bl = img[2 * HWSZ + idx];
    float dval = mu0[0] + mu1[0] * fmaxf(g, bl) + mu2[0] * r;
    float D0 = 0.f, D1 = 0.f, D2 = 0.f;
    if (sc >= 0) { D0 = sD[sc * 3 + 0]; D1 = sD[sc * 3 + 1]; D2 = sD[sc * 3 + 2]; }
    float inv = 1.0f / cnt;  // center neighbor always valid -> cnt >= 1
    out[idx] = (D0 + n0 * inv) * dval;
    out[HWSZ + idx] = (D1 + n1 * inv) * dval;
    out[2 * HWSZ + idx] = (D2 + n2 * inv) * dval;
}

extern "C" void kernel_launch(void* const* d_in, const int* in_sizes, int n_in,
                              void* d_out, int out_size, void* d_ws, size_t ws_size,
                              hipStream_t stream) {
    const float* img = (const float*)d_in[0];
    const float* depth = (const float*)d_in[1];
    const float* mu0 = (const float*)d_in[2];
    const float* mu1 = (const float*)d_in[3];
    const float* mu2 = (const float*)d_in[4];
    float* out = (float*)d_out;
    unsigned int* ws = (unsigned int*)d_ws;

    eigm_init<<<(W_TOTAL + 255) / 256, 256, 0, stream>>>(ws);
    eigm_minmax<<<1024, 256, 0, stream>>>(depth, ws);
    eigm_accum<<<512, 256, 0, stream>>>(img, depth, ws);
    eigm_stats<<<1, 32, 0, stream>>>(ws);
    eigm_final<<<dim3(Wdim / 32, Hdim / 8), dim3(32, 8), 0, stream>>>(
        img, depth, mu0, mu1, mu2, (const float*)ws, out);
}